// PointWiseGlobalFusion_87789131530948
// MI455X (gfx1250) — compile-verified
//
#include <hip/hip_runtime.h>

typedef float v2f __attribute__((ext_vector_type(2)));
typedef float v8f __attribute__((ext_vector_type(8)));

// Problem constants (from reference setup_inputs)
constexpr int BB    = 2;      // batch
constexpr int NF    = 5;      // frames
constexpr int CC    = 64;     // channels
constexpr int HH    = 256;
constexpr int WW    = 256;
constexpr int GG    = 4;      // groups
constexpr int CPG   = 16;     // channels per group
constexpr int KTOT  = 144;    // 16 ic * 9 taps
constexpr int PLANE = HH * WW;

constexpr int WTILE = 64;     // pixels per block (4 wmma tiles)
constexpr int TPB   = 128;    // 4 wave32s
constexpr int SICOL = WTILE + 2;          // 66 (halo cols)
constexpr int SIROW = 3 * SICOL;          // 198
constexpr int SIN_N = CPG * SIROW;        // 3168 floats

__global__ __launch_bounds__(TPB) void nlfuse_wmma_kernel(
    const float* __restrict__ X,  const float* __restrict__ W1,
    const float* __restrict__ B1, const float* __restrict__ W2,
    const float* __restrict__ B2, float* __restrict__ OUT)
{
    __shared__ float sIn[SIN_N];                    // input halo, one frame
    __shared__ float sEmb[2 * NF * CPG * WTILE];    // emb1/emb2 for all frames

    // ---- block decode: (b, g, h, wblk) ----
    int bid  = blockIdx.x;
    int wblk = bid % (WW / WTILE);   bid /= (WW / WTILE);
    int h    = bid % HH;             bid /= HH;
    int g    = bid % GG;
    int b    = bid / GG;
    const int w0 = wblk * WTILE;

    const int t     = threadIdx.x;
    const int lane  = t & 31;
    const int wave  = t >> 5;
    const int khalf = lane >> 4;     // selects K pair (A) / K pair (B)
    const int mrow  = lane & 15;     // M row for A, pixel col for B/D
    const int tilebase = wave * 16;

    // per-lane weight row pointers: w[oc][ic][ky][kx], oc = g*16 + mrow
    const float* w1p = W1 + (g * CPG + mrow) * KTOT;
    const float* w2p = W2 + (g * CPG + mrow) * KTOT;

    // =========================== Phase 1: grouped convs via WMMA ===========================
    for (int f = 0; f < NF; ++f) {
        __syncthreads();   // previous frame's sIn reads done
        const float* xsrc = X + ((size_t)(b * NF + f) * CC + g * CPG) * PLANE;
        for (int idx = t; idx < SIN_N; idx += TPB) {
            int ic  = idx / SIROW;
            int rem = idx - ic * SIROW;
            int r   = rem / SICOL;
            int cc  = rem - r * SICOL;
            int gr  = h + r - 1;
            int gc  = w0 + cc - 1;
            float v = 0.f;
            if ((unsigned)gr < (unsigned)HH && (unsigned)gc < (unsigned)WW)
                v = xsrc[ic * PLANE + gr * WW + gc];
            sIn[idx] = v;
        }
        // hint: prefetch next frame's rows (gfx1250 global_prefetch_b8)
        if (f + 1 < NF) {
            const float* nx = X + ((size_t)(b * NF + f + 1) * CC + g * CPG) * PLANE
                              + h * WW + w0;
            __builtin_prefetch(nx + (size_t)(t & 15) * PLANE, 0, 1);
        }
        __syncthreads();

        // implicit GEMM: D[16 oc x 16 px] += A[16 x 4] * B[4 x 16], K = 144, tap-major k
        v8f acc1 = {0.f,0.f,0.f,0.f,0.f,0.f,0.f,0.f};
        v8f acc2 = {0.f,0.f,0.f,0.f,0.f,0.f,0.f,0.f};
        #pragma unroll 4
        for (int ks = 0; ks < KTOT / 4; ++ks) {
            int k0  = ks * 4 + khalf * 2;       // even; k1 = k0+1 shares the same tap
            int tap = k0 >> 4;                   // 0..8
            int ic0 = k0 & 15;                   // even
            int dy  = tap / 3;
            int dx  = tap - dy * 3;
            int bcol = tilebase + mrow + dx;     // halo-local pixel column
            int sidx = ic0 * SIROW + dy * SICOL + bcol;
            v2f bv;  bv.x = sIn[sidx];           bv.y = sIn[sidx + SIROW];
            int wk   = ic0 * 9 + tap;
            v2f a1;  a1.x = w1p[wk];             a1.y = w1p[wk + 9];
            v2f a2;  a2.x = w2p[wk];             a2.y = w2p[wk + 9];
            acc1 = __builtin_amdgcn_wmma_f32_16x16x4_f32(
                       false, a1, false, bv, (short)0, acc1, false, false);
            acc2 = __builtin_amdgcn_wmma_f32_16x16x4_f32(
                       false, a2, false, bv, (short)0, acc2, false, false);
        }
        // epilogue: bias add, stash embeddings in LDS
        #pragma unroll
        for (int v = 0; v < 8; ++v) {
            int M = v + khalf * 8;
            float e1 = acc1[v] + B1[g * CPG + M];
            float e2 = acc2[v] + B2[g * CPG + M];
            sEmb[((0 * NF + f) * CPG + M) * WTILE + tilebase + mrow] = e1;
            sEmb[((1 * NF + f) * CPG + M) * WTILE + tilebase + mrow] = e2;
        }
    }
    __syncthreads();

    // =========================== Phase 2: affinity + softmax + fuse ===========================
    const int p  = t & (WTILE - 1);       // pixel within strip
    const int c0 = (t >> 6) * 8;          // channel half within group
    const int hw = h * WW + w0 + p;

    // cache emb1 rows for this pixel
    float e1r[NF][CPG];
    #pragma unroll
    for (int n = 0; n < NF; ++n)
        #pragma unroll
        for (int c = 0; c < CPG; ++c)
            e1r[n][c] = sEmb[((0 * NF + n) * CPG + c) * WTILE + p];

    // cor[n][m] = <emb1[n], emb2[m]> over 16 embedding channels
    float att[NF][NF];
    #pragma unroll
    for (int mm = 0; mm < NF; ++mm) {
        float e2c[CPG];
        #pragma unroll
        for (int c = 0; c < CPG; ++c)
            e2c[c] = sEmb[((1 * NF + mm) * CPG + c) * WTILE + p];
        #pragma unroll
        for (int n = 0; n < NF; ++n) {
            float s = 0.f;
            #pragma unroll
            for (int c = 0; c < CPG; ++c) s += e1r[n][c] * e2c[c];
            att[n][mm] = s;
        }
    }
    // softmax over source frame mm
    #pragma unroll
    for (int n = 0; n < NF; ++n) {
        float mx = att[n][0];
        #pragma unroll
        for (int mm = 1; mm < NF; ++mm) mx = fmaxf(mx, att[n][mm]);
        float s = 0.f;
        #pragma unroll
        for (int mm = 0; mm < NF; ++mm) {
            float ev = __expf(att[n][mm] - mx);
            att[n][mm] = ev;
            s += ev;
        }
        float inv = 1.0f / s;
        #pragma unroll
        for (int mm = 0; mm < NF; ++mm) att[n][mm] *= inv;
    }

    // fuse: passthrough x, nl_feat = att @ x, frame-max pool; this block owns
    // 16 channels of each of the 3 output sections -> no duplicated stores.
    #pragma unroll
    for (int c = 0; c < 8; ++c) {
        int ch = g * CPG + c0 + c;
        float xv[NF];
        #pragma unroll
        for (int mm = 0; mm < NF; ++mm)
            xv[mm] = X[((size_t)(b * NF + mm) * CC + ch) * PLANE + hw];
        float pool = xv[0];
        #pragma unroll
        for (int mm = 1; mm < NF; ++mm) pool = fmaxf(pool, xv[mm]);
        #pragma unroll
        for (int n = 0; n < NF; ++n) {
            float nl = 0.f;
            #pragma unroll
            for (int mm = 0; mm < NF; ++mm) nl += att[n][mm] * xv[mm];
            size_t ob = (size_t)(b * NF + n) * (3 * CC) * PLANE + hw;
            OUT[ob + (size_t)ch              * PLANE] = xv[n];  // section 0: x
            OUT[ob + (size_t)(CC + ch)       * PLANE] = nl;     // section 1: nl_feat
            OUT[ob + (size_t)(2 * CC + ch)   * PLANE] = pool;   // section 2: pool
        }
    }
}

extern "C" void kernel_launch(void* const* d_in, const int* in_sizes, int n_in,
                              void* d_out, int out_size, void* d_ws, size_t ws_size,
                              hipStream_t stream) {
    (void)in_sizes; (void)n_in; (void)d_ws; (void)ws_size; (void)out_size;
    const float* x  = (const float*)d_in[0];
    const float* w1 = (const float*)d_in[1];
    const float* b1 = (const float*)d_in[2];
    const float* w2 = (const float*)d_in[3];
    const float* b2 = (const float*)d_in[4];
    float* out = (float*)d_out;

    dim3 grid(BB * GG * HH * (WW / WTILE));   // 8192 blocks
    nlfuse_wmma_kernel<<<grid, TPB, 0, stream>>>(x, w1, b1, w2, b2, out);
}